// FDSCS_34205119545973
// MI455X (gfx1250) — compile-verified
//
#include <hip/hip_runtime.h>
#include <stdint.h>

// Problem geometry (fixed by the reference harness).
#define H 320
#define W 960
#define P (H * W)          // one plane = 307200 elements
#define TILE 320           // columns per block in the volume kernel (10 wave32s)
#define LSPAN 448          // TILE + maxdisp(128) left-operand span staged in LDS

// ---------------------------------------------------------------------------
// Workspace plane layout (all float-sized slots, some hold int32 census):
//   0: census_L (i32)   1: Cb_L   2: Cr_L
//   3: census_R (i32)   4: Cb_R   5: Cr_R
//   6: Y_L              7: Y_R
// Left planes (0..2) come first so the b128 async over-read at the right edge
// of the last x-tile spills into plane 3, which is still inside the workspace.
// Total: 8 * P * 4 B = 9.83 MB.
// ---------------------------------------------------------------------------

__global__ __launch_bounds__(256)
void fdscs_ycbcr_kernel(const float* __restrict__ left,
                        const float* __restrict__ right,
                        float* __restrict__ ws) {
    int idx = blockIdx.x * 256 + threadIdx.x;   // 0 .. 2P-1 (pixel of one image)
    if (idx >= 2 * P) return;
    int img = (idx >= P) ? 1 : 0;
    int p = idx - img * P;
    const float* src = img ? right : left;
    float r = src[p];
    float g = src[P + p];
    float b = src[2 * P + p];
    float y  = 0.299f * r + 0.587f * g + 0.114f * b;
    float cb = (b - y) * 0.564f + 0.5f;
    float cr = (r - y) * 0.713f + 0.5f;
    ws[(size_t)(img ? 7 : 6) * P + p] = y;
    ws[(size_t)(img ? 4 : 1) * P + p] = cb;
    ws[(size_t)(img ? 5 : 2) * P + p] = cr;
}

__global__ __launch_bounds__(256)
void fdscs_census_kernel(float* __restrict__ ws) {
    int idx = blockIdx.x * 256 + threadIdx.x;   // 0 .. 2P-1
    if (idx >= 2 * P) return;
    int img = (idx >= P) ? 1 : 0;
    int p = idx - img * P;
    const float* Y = ws + (size_t)(img ? 7 : 6) * P;
    int* dst = ((int*)ws) + (size_t)(img ? 3 : 0) * P;
    int yy = p / W;
    int xx = p - yy * W;
    int c = 0;
    if (yy >= 2 && yy < H - 2 && xx >= 2 && xx < W - 2) {
        float cen = Y[p];
        const float* base = Y + (yy - 2) * W + (xx - 2);
        #pragma unroll
        for (int v = 0; v < 5; ++v) {
            #pragma unroll
            for (int u = 0; u < 5; ++u) {
                if (u == 1 && v == 1) continue;     // reference skips (u==1 && v==1)
                c = (c << 1) | ((base[v * W + u] >= cen) ? 1 : 0);
            }
        }
    }
    dst[p] = c;   // zero-padded border, matching jnp.pad
}

// One block = one image row x one 320-column tile x all `maxdisp` disparities.
// Left operands for columns [x0, x0+447] are staged into LDS with gfx1250
// async load-to-LDS (each value is reused up to 128x across disparities).
__global__ __launch_bounds__(TILE)
void fdscs_volume_kernel(const float* __restrict__ wsf,
                         float* __restrict__ out,
                         int maxdisp) {
    __shared__ __align__(16) int   s_cl[LSPAN];
    __shared__ __align__(16) float s_cb[LSPAN];
    __shared__ __align__(16) float s_cr[LSPAN];

    const int*   cl   = (const int*)  (wsf + 0 * (size_t)P);
    const float* cbl  =                wsf + 1 * (size_t)P;
    const float* crl  =                wsf + 2 * (size_t)P;
    const int*   cenr = (const int*)  (wsf + 3 * (size_t)P);
    const float* cbr  =                wsf + 4 * (size_t)P;
    const float* crr  =                wsf + 5 * (size_t)P;

    const int tid = threadIdx.x;
    const int x0  = blockIdx.x * TILE;
    const int y   = blockIdx.y;
    const int row = y * W;

    // --- Async stage of left operands: 112 lanes x b128 covers 448 dwords per
    // array. Global and LDS addresses are 16B aligned (x0, row, tid*16 all
    // multiples of 16B). Tracked by ASYNCcnt.
    if (tid < LSPAN / 4) {
        const int e = row + x0 + tid * 4;
        unsigned long long g0 = (unsigned long long)(uintptr_t)(cl  + e);
        unsigned long long g1 = (unsigned long long)(uintptr_t)(cbl + e);
        unsigned long long g2 = (unsigned long long)(uintptr_t)(crl + e);
        // Low 32 bits of a flat LDS address are the wave-relative LDS offset.
        unsigned l0 = (unsigned)(uintptr_t)(&s_cl[tid * 4]);
        unsigned l1 = (unsigned)(uintptr_t)(&s_cb[tid * 4]);
        unsigned l2 = (unsigned)(uintptr_t)(&s_cr[tid * 4]);
        asm volatile("global_load_async_to_lds_b128 %0, %1, off\n\t"
                     "global_load_async_to_lds_b128 %2, %3, off\n\t"
                     "global_load_async_to_lds_b128 %4, %5, off"
                     :
                     : "v"(l0), "v"(g0), "v"(l1), "v"(g1), "v"(l2), "v"(g2)
                     : "memory");
    }

    // Right-image operands stay in registers (overlaps with the async fill).
    const int   x    = x0 + tid;
    const int   rcen = cenr[row + x];
    const float rcb  = cbr[row + x];
    const float rcr  = crr[row + x];

#if __has_builtin(__builtin_amdgcn_s_wait_asynccnt)
    __builtin_amdgcn_s_wait_asynccnt(0);
#else
    asm volatile("s_wait_asynccnt 0" ::: "memory");
#endif
    __syncthreads();

    float* o1 = out + row + x;                    // Hamming volume, planes [0, D)
    float* o2 = o1 + (size_t)maxdisp * P;         // |dCb| volume,   planes [D, 2D)
    float* o3 = o2 + (size_t)maxdisp * P;         // |dCr| volume,   planes [2D, 3D)
    const int dvalid = W - x;                     // d < dvalid  <=>  x + d < W

    #pragma unroll 4
    for (int d = 0; d < maxdisp; ++d) {
        const bool ok = d < dvalid;
        float v1 = ok ? (float)__popc(s_cl[tid + d] ^ rcen) : 0.0f;
        float v2 = ok ? fabsf(s_cb[tid + d] - rcb) : 0.0f;
        float v3 = ok ? fabsf(s_cr[tid + d] - rcr) : 0.0f;
        const size_t off = (size_t)d * P;
        // 472 MB write stream: non-temporal so it doesn't thrash L2.
        __builtin_nontemporal_store(v1, o1 + off);
        __builtin_nontemporal_store(v2, o2 + off);
        __builtin_nontemporal_store(v3, o3 + off);
    }
}

extern "C" void kernel_launch(void* const* d_in, const int* in_sizes, int n_in,
                              void* d_out, int out_size, void* d_ws, size_t ws_size,
                              hipStream_t stream) {
    const float* left  = (const float*)d_in[0];
    const float* right = (const float*)d_in[1];
    float* out = (float*)d_out;
    float* wsf = (float*)d_ws;

    // maxdisp derivable on the host from out_size (no device sync; capture-safe).
    int maxdisp = out_size / (3 * H * W);         // == 128 for this harness
    if (maxdisp > 128) maxdisp = 128;             // LDS span guard

    const int n2p = 2 * P;
    fdscs_ycbcr_kernel <<<(n2p + 255) / 256, 256, 0, stream>>>(left, right, wsf);
    fdscs_census_kernel<<<(n2p + 255) / 256, 256, 0, stream>>>(wsf);

    dim3 grid(W / TILE, H);                       // (3, 320) = 960 blocks
    fdscs_volume_kernel<<<grid, TILE, 0, stream>>>(wsf, out, maxdisp);
}